// USEF_TSE_Mamba_v7_Enhanced_36447092474496
// MI455X (gfx1250) — compile-verified
//
#include <hip/hip_runtime.h>
#include <cstdint>
#include <cstddef>

// ============================================================================
// USEF-TSE Mamba forward for gfx1250 (MI455X).
// fp32 everywhere; all GEMMs via V_WMMA_F32_16X16X4_F32 (full-precision WMMA).
// Layout convention: token-major (B, L, C), idx = (b*L + l)*C + c.
// ============================================================================

typedef float v2f __attribute__((ext_vector_type(2)));
typedef float v8f __attribute__((ext_vector_type(8)));

#define DEVINL __device__ __forceinline__

DEVINL float dev_sigmoid(float x) { return 1.f / (1.f + __expf(-x)); }
DEVINL float dev_silu(float x)    { return x * dev_sigmoid(x); }

// ---------------------------------------------------------------------------
// WMMA GEMM: Y[M,N] = act( X[M,K] * op(W) + bias[N] )
//   TRANSB=true : op(W)=W^T with W stored [N,K] (PyTorch Linear weight)
//   TRANSB=false: op(W)=W   with W stored [K,N] (row stride ldb)
//   VEC=true    : fragment loads as float2 (global_load_b64); host guarantees
//                 even strides + 8B-aligned bases on the vectorized operands.
//
// Per wave: one 16(M) x 32(N) strip = 2 accumulators sharing one A fragment.
// 4 waves per block -> 32(M) x 64(N) block tile.
//
// Wave-uniform decisions (active / interior) are forced into SGPRs via
// readfirstlane so the compiler emits scalar branches, not EXEC save/restore.
// Interior tiles run a pure load->wmma loop with no masking VALU; edge tiles
// use clamped indices + 0/1 mask multiplies (branch-free).  WMMA therefore
// always executes with full EXEC (ISA 7.12 requirement).
// ---------------------------------------------------------------------------
template <bool TRANSB, bool VEC>
__global__ void k_gemm(const float* __restrict__ X, int lda,
                       const float* __restrict__ W, int ldb,
                       float* __restrict__ Y, int ldc,
                       const float* __restrict__ bias,
                       int M, int N, int K, int act)
{
    const int lane = threadIdx.x & 31;
    const int wave = threadIdx.x >> 5;
    const int row0 = blockIdx.y * 32 + (wave >> 1) * 16;
    const int col0 = blockIdx.x * 64 + (wave & 1) * 32;

    const int active = __builtin_amdgcn_readfirstlane((row0 < M && col0 < N) ? 1 : 0);
    if (!active) return;
    const int interior = __builtin_amdgcn_readfirstlane(
        ((row0 + 16 <= M) && (col0 + 32 <= N)) ? 1 : 0);

    const int l15  = lane & 15;
    const int koff = (lane >> 4) << 1;       // lanes 0-15 -> K+0,1 ; 16-31 -> K+2,3

    // clamped indices + 0/1 masks (used only on edge tiles)
    const int   mrowR = row0 + l15;
    const float am    = (mrowR < M) ? 1.f : 0.f;
    const int   mrow  = (mrowR < M) ? mrowR : (M - 1);

    const int   nc0R  = col0 + l15;
    const float bm0   = (nc0R < N) ? 1.f : 0.f;
    const int   nc0   = (nc0R < N) ? nc0R : (N - 1);
    const int   nc1R  = nc0R + 16;
    const float bm1   = (nc1R < N) ? 1.f : 0.f;
    const int   nc1   = (nc1R < N) ? nc1R : (N - 1);

    const float* Xp  = X + (size_t)mrow * lda + koff;
    const float* Wp0 = TRANSB ? (W + (size_t)nc0 * ldb + koff) : (W + nc0);
    const float* Wp1 = TRANSB ? (W + (size_t)nc1 * ldb + koff) : (W + nc1);

    v8f acc0 = {0.f, 0.f, 0.f, 0.f, 0.f, 0.f, 0.f, 0.f};
    v8f acc1 = {0.f, 0.f, 0.f, 0.f, 0.f, 0.f, 0.f, 0.f};

    const int Kmain = K & ~3;

    if (interior) {
        // ---- hot path: no masks, no branches: load -> wmma ----
#pragma unroll 4
        for (int k0 = 0; k0 < Kmain; k0 += 4) {
            v2f a, b0, b1;
            if (VEC) {
                a = *(const v2f*)(Xp + k0);
            } else {
                a.x = Xp[k0]; a.y = Xp[k0 + 1];
            }
            if (TRANSB) {
                if (VEC) {
                    b0 = *(const v2f*)(Wp0 + k0);
                    b1 = *(const v2f*)(Wp1 + k0);
                } else {
                    b0.x = Wp0[k0]; b0.y = Wp0[k0 + 1];
                    b1.x = Wp1[k0]; b1.y = Wp1[k0 + 1];
                }
            } else {
                const size_t kk = (size_t)(k0 + koff);
                b0.x = Wp0[kk * ldb]; b0.y = Wp0[(kk + 1) * ldb];
                b1.x = Wp1[kk * ldb]; b1.y = Wp1[(kk + 1) * ldb];
            }
            acc0 = __builtin_amdgcn_wmma_f32_16x16x4_f32(false, a, false, b0,
                                                         (short)0, acc0, false, false);
            acc1 = __builtin_amdgcn_wmma_f32_16x16x4_f32(false, a, false, b1,
                                                         (short)0, acc1, false, false);
        }
    } else {
        // ---- edge path: clamped loads + 0/1 mask multiplies, still branch-free
#pragma unroll 2
        for (int k0 = 0; k0 < Kmain; k0 += 4) {
            v2f a, b0, b1;
            if (VEC) {
                a = *(const v2f*)(Xp + k0);
            } else {
                a.x = Xp[k0]; a.y = Xp[k0 + 1];
            }
            if (TRANSB) {
                if (VEC) {
                    b0 = *(const v2f*)(Wp0 + k0);
                    b1 = *(const v2f*)(Wp1 + k0);
                } else {
                    b0.x = Wp0[k0]; b0.y = Wp0[k0 + 1];
                    b1.x = Wp1[k0]; b1.y = Wp1[k0 + 1];
                }
            } else {
                const size_t kk = (size_t)(k0 + koff);
                b0.x = Wp0[kk * ldb]; b0.y = Wp0[(kk + 1) * ldb];
                b1.x = Wp1[kk * ldb]; b1.y = Wp1[(kk + 1) * ldb];
            }
            a.x *= am;  a.y *= am;
            b0.x *= bm0; b0.y *= bm0;
            b1.x *= bm1; b1.y *= bm1;
            acc0 = __builtin_amdgcn_wmma_f32_16x16x4_f32(false, a, false, b0,
                                                         (short)0, acc0, false, false);
            acc1 = __builtin_amdgcn_wmma_f32_16x16x4_f32(false, a, false, b1,
                                                         (short)0, acc1, false, false);
        }
    }

    if (Kmain < K) {   // masked K tail (clamped loads stay in-bounds)
        const int   kk  = Kmain + koff;
        const float m0  = (kk < K) ? 1.f : 0.f;
        const float m1  = (kk + 1 < K) ? 1.f : 0.f;
        const int   kc0 = (kk < K) ? kk : (K - 1);
        const int   kc1 = (kk + 1 < K) ? (kk + 1) : (K - 1);
        v2f a, b0, b1;
        a.x = X[(size_t)mrow * lda + kc0] * m0;
        a.y = X[(size_t)mrow * lda + kc1] * m1;
        if (TRANSB) {
            b0.x = W[(size_t)nc0 * ldb + kc0] * m0; b0.y = W[(size_t)nc0 * ldb + kc1] * m1;
            b1.x = W[(size_t)nc1 * ldb + kc0] * m0; b1.y = W[(size_t)nc1 * ldb + kc1] * m1;
        } else {
            b0.x = W[(size_t)kc0 * ldb + nc0] * m0; b0.y = W[(size_t)kc1 * ldb + nc0] * m1;
            b1.x = W[(size_t)kc0 * ldb + nc1] * m0; b1.y = W[(size_t)kc1 * ldb + nc1] * m1;
        }
        a.x *= am;  a.y *= am;
        b0.x *= bm0; b0.y *= bm0;
        b1.x *= bm1; b1.y *= bm1;
        acc0 = __builtin_amdgcn_wmma_f32_16x16x4_f32(false, a, false, b0,
                                                     (short)0, acc0, false, false);
        acc1 = __builtin_amdgcn_wmma_f32_16x16x4_f32(false, a, false, b1,
                                                     (short)0, acc1, false, false);
    }

    // ---- store (C/D layout: VGPR r, lanes 0-15 -> M=r, lanes 16-31 -> M=r+8)
    const int  rbase = row0 + ((lane >> 4) << 3);
    const bool okc0  = (nc0R < N);
    const bool okc1  = (nc1R < N);
    const float bv0  = bias ? bias[nc0] : 0.f;
    const float bv1  = bias ? bias[nc1] : 0.f;
#pragma unroll
    for (int r = 0; r < 8; ++r) {
        const int m = rbase + r;
        if (m < M) {
            if (okc0) {
                float v = acc0[r] + bv0;
                if (act == 1)      v = dev_silu(v);
                else if (act == 2) v = dev_sigmoid(v);
                Y[(size_t)m * ldc + nc0R] = v;
            }
            if (okc1) {
                float v = acc1[r] + bv1;
                if (act == 1)      v = dev_silu(v);
                else if (act == 2) v = dev_sigmoid(v);
                Y[(size_t)m * ldc + nc1R] = v;
            }
        }
    }
}

// ---------------------------------------------------------------------------
// Strided conv (K=16, S=2, pad=8) fused with BN + SiLU.  x:(B,Lin,Cin) -> y:(B,Lout,Cout)
// ---------------------------------------------------------------------------
__global__ void k_conv_down(const float* __restrict__ x, int Lin, int Cin,
                            const float* __restrict__ w, const float* __restrict__ bias,
                            const float* __restrict__ bg, const float* __restrict__ bb,
                            const float* __restrict__ bm, const float* __restrict__ bv,
                            float* __restrict__ y, int Lout, int Cout, int Bn)
{
    long long idx = (long long)blockIdx.x * blockDim.x + threadIdx.x;
    long long total = (long long)Bn * Lout * Cout;
    if (idx >= total) return;
    const int o = (int)(idx % Cout);
    long long t = idx / Cout;
    const int l = (int)(t % Lout);
    const int b = (int)(t / Lout);

    float acc = bias[o];
    const int base = 2 * l - 8;
    for (int i = 0; i < Cin; ++i) {
        const float* wr = w + ((size_t)o * Cin + i) * 16;
        for (int k = 0; k < 16; ++k) {
            const int ti = base + k;
            if (ti >= 0 && ti < Lin)
                acc += wr[k] * x[((size_t)b * Lin + ti) * Cin + i];
        }
    }
    const float sc = bg[o] * rsqrtf(bv[o] + 1e-5f);
    const float v = acc * sc + (bb[o] - bm[o] * sc);
    y[((size_t)b * Lout + l) * Cout + o] = dev_silu(v);
}

// ---------------------------------------------------------------------------
// ConvTranspose (K=16, S=2, pad=8, out_pad=1), optionally fused BN+SiLU.
// w layout (Cin, Cout, 16).  y[l] = sum_k w[i,o,k]*x[(l+8-k)/2]  for even l+8-k.
// ---------------------------------------------------------------------------
__global__ void k_convT(const float* __restrict__ x, int Lin, int Cin,
                        const float* __restrict__ w, const float* __restrict__ bias,
                        const float* __restrict__ bg, const float* __restrict__ bb,
                        const float* __restrict__ bm, const float* __restrict__ bv,
                        int use_bn,
                        float* __restrict__ y, int Lout, int Cout, int Bn)
{
    long long idx = (long long)blockIdx.x * blockDim.x + threadIdx.x;
    long long total = (long long)Bn * Lout * Cout;
    if (idx >= total) return;
    const int o = (int)(idx % Cout);
    long long t = idx / Cout;
    const int l = (int)(t % Lout);
    const int b = (int)(t / Lout);

    float acc = bias[o];
    for (int k = 0; k < 16; ++k) {
        const int p = l + 8 - k;
        if (p >= 0 && (p & 1) == 0) {
            const int ti = p >> 1;
            if (ti < Lin) {
                const float* xr = x + ((size_t)b * Lin + ti) * Cin;
                for (int i = 0; i < Cin; ++i)
                    acc += w[((size_t)i * Cout + o) * 16 + k] * xr[i];
            }
        }
    }
    if (use_bn) {
        const float sc = bg[o] * rsqrtf(bv[o] + 1e-5f);
        acc = dev_silu(acc * sc + (bb[o] - bm[o] * sc));
    }
    y[((size_t)b * Lout + l) * Cout + o] = acc;
}

// ---------------------------------------------------------------------------
// LayerNorm over last dim C (one block / token).  act: 0=none, 1=silu
// ---------------------------------------------------------------------------
__global__ void k_layernorm(const float* __restrict__ x, float* __restrict__ y,
                            const float* __restrict__ g, const float* __restrict__ b,
                            int C, int act)
{
    __shared__ float red[256];
    const size_t t = blockIdx.x;
    const float* xr = x + t * C;
    float* yr = y + t * C;

    float s = 0.f;
    for (int c = threadIdx.x; c < C; c += 256) s += xr[c];
    red[threadIdx.x] = s; __syncthreads();
    for (int st = 128; st > 0; st >>= 1) {
        if ((int)threadIdx.x < st) red[threadIdx.x] += red[threadIdx.x + st];
        __syncthreads();
    }
    const float mean = red[0] / (float)C;
    __syncthreads();

    float v = 0.f;
    for (int c = threadIdx.x; c < C; c += 256) { float d = xr[c] - mean; v += d * d; }
    red[threadIdx.x] = v; __syncthreads();
    for (int st = 128; st > 0; st >>= 1) {
        if ((int)threadIdx.x < st) red[threadIdx.x] += red[threadIdx.x + st];
        __syncthreads();
    }
    const float rstd = rsqrtf(red[0] / (float)C + 1e-5f);

    for (int c = threadIdx.x; c < C; c += 256) {
        float val = (xr[c] - mean) * rstd * g[c] + b[c];
        if (act == 1) val = dev_silu(val);
        yr[c] = val;
    }
}

// ---------------------------------------------------------------------------
// BatchNorm (per-channel affine) elementwise, optional SiLU.
// ---------------------------------------------------------------------------
__global__ void k_bn_act(const float* __restrict__ x, float* __restrict__ y,
                         const float* __restrict__ g, const float* __restrict__ b,
                         const float* __restrict__ m, const float* __restrict__ v,
                         int C, long long n, int act)
{
    long long i = (long long)blockIdx.x * blockDim.x + threadIdx.x;
    if (i >= n) return;
    const int c = (int)(i % C);
    const float sc = g[c] * rsqrtf(v[c] + 1e-5f);
    float val = x[i] * sc + (b[c] - m[c] * sc);
    if (act) val = dev_silu(val);
    y[i] = val;
}

// ---------------------------------------------------------------------------
// Mamba depthwise causal conv (DC=4, left pad 3) + bias + SiLU.
// Input = xi view into xz (row stride `stride` = 2*di).
// ---------------------------------------------------------------------------
__global__ void k_dwconv_silu(const float* __restrict__ xz, int stride,
                              const float* __restrict__ w, const float* __restrict__ bias,
                              float* __restrict__ xc, int L, int di, int Bn)
{
    long long i = (long long)blockIdx.x * blockDim.x + threadIdx.x;
    long long n = (long long)Bn * L * di;
    if (i >= n) return;
    const int c = (int)(i % di);
    long long t = i / di;
    const int l = (int)(t % L);
    const int b = (int)(t / L);

    float acc = bias[c];
#pragma unroll
    for (int j = 0; j < 4; ++j) {
        const int li = l - 3 + j;
        if (li >= 0)
            acc += w[c * 4 + j] * xz[((size_t)b * L + li) * stride + c];
    }
    xc[((size_t)b * L + l) * di + c] = dev_silu(acc);
}

// ---------------------------------------------------------------------------
// Mamba selective scan. One block per batch, one thread per inner channel d.
// 16-wide SSM state lives in registers; shared B/C broadcast via LDS.
// ---------------------------------------------------------------------------
__global__ void k_scan(const float* __restrict__ dtpre,  // (B*L, di), pre-softplus
                       const float* __restrict__ xdbl,   // (B*L, ndbl): B at +dr, C at +dr+16
                       const float* __restrict__ xc,     // (B*L, di)
                       const float* __restrict__ A_log,  // (di,16)
                       const float* __restrict__ Dp,     // (di)
                       float* __restrict__ y,            // (B*L, di)
                       int L, int di, int ndbl, int dr)
{
    const int b = blockIdx.x;
    const int d = threadIdx.x;
    __shared__ float sB[16];
    __shared__ float sC[16];

    float a[16], h[16];
#pragma unroll
    for (int s = 0; s < 16; ++s) {
        a[s] = -__expf(A_log[(size_t)d * 16 + s]);
        h[s] = 0.f;
    }
    const float Dv = Dp[d];

    for (int l = 0; l < L; ++l) {
        const size_t t = (size_t)b * L + l;
        if (d < 16)       sB[d]      = xdbl[t * ndbl + dr + d];
        else if (d < 32)  sC[d - 16] = xdbl[t * ndbl + dr + d];
        __syncthreads();

        float dt = dtpre[t * di + d];
        dt = (dt > 20.f) ? dt : log1pf(__expf(dt));     // softplus
        const float xv = xc[t * di + d];
        const float dtx = dt * xv;
        float acc = 0.f;
#pragma unroll
        for (int s = 0; s < 16; ++s) {
            h[s] = h[s] * __expf(dt * a[s]) + dtx * sB[s];
            acc += h[s] * sC[s];
        }
        y[t * di + d] = acc + xv * Dv;
        __syncthreads();
    }
}

// ---------------------------------------------------------------------------
// Row softmax with scale + clip(+-50) (attention). One block per row.
// ---------------------------------------------------------------------------
__global__ void k_softmax_row(float* __restrict__ s, int ncols, float scale)
{
    __shared__ float red[256];
    float* r = s + (size_t)blockIdx.x * ncols;

    float mx = -1e30f;
    for (int c = threadIdx.x; c < ncols; c += 256) {
        float v = fminf(50.f, fmaxf(-50.f, r[c] * scale));
        mx = fmaxf(mx, v);
    }
    red[threadIdx.x] = mx; __syncthreads();
    for (int st = 128; st > 0; st >>= 1) {
        if ((int)threadIdx.x < st) red[threadIdx.x] = fmaxf(red[threadIdx.x], red[threadIdx.x + st]);
        __syncthreads();
    }
    const float rowmax = red[0]; __syncthreads();

    float sum = 0.f;
    for (int c = threadIdx.x; c < ncols; c += 256) {
        float v = fminf(50.f, fmaxf(-50.f, r[c] * scale));
        float e = __expf(v - rowmax);
        r[c] = e; sum += e;
    }
    red[threadIdx.x] = sum; __syncthreads();
    for (int st = 128; st > 0; st >>= 1) {
        if ((int)threadIdx.x < st) red[threadIdx.x] += red[threadIdx.x + st];
        __syncthreads();
    }
    const float rs = 1.f / red[0];
    for (int c = threadIdx.x; c < ncols; c += 256) r[c] *= rs;
}

// ------------------------- small elementwise helpers -----------------------
__global__ void k_copy(const float* __restrict__ a, float* __restrict__ y, long long n)
{ long long i = (long long)blockIdx.x * blockDim.x + threadIdx.x; if (i < n) y[i] = a[i]; }

__global__ void k_add(const float* __restrict__ a, const float* __restrict__ b,
                      float* __restrict__ y, long long n)
{ long long i = (long long)blockIdx.x * blockDim.x + threadIdx.x; if (i < n) y[i] = a[i] + b[i]; }

__global__ void k_pos_add(const float* __restrict__ x, const float* __restrict__ pos,
                          float* __restrict__ y, int L, int C, long long n)
{
    long long i = (long long)blockIdx.x * blockDim.x + threadIdx.x;
    if (i >= n) return;
    const int c = (int)(i % C);
    const int l = (int)((i / C) % L);
    y[i] = x[i] + pos[(size_t)l * C + c];
}

__global__ void k_concat2(const float* __restrict__ a, int Ca,
                          const float* __restrict__ b, int Cb,
                          float* __restrict__ y, long long ntok)
{
    long long n = ntok * (Ca + Cb);
    long long i = (long long)blockIdx.x * blockDim.x + threadIdx.x;
    if (i >= n) return;
    const int c = (int)(i % (Ca + Cb));
    const long long t = i / (Ca + Cb);
    y[i] = (c < Ca) ? a[t * Ca + c] : b[t * Cb + (c - Ca)];
}

__global__ void k_mul_silu(const float* __restrict__ y_in, const float* __restrict__ xz,
                           int di, float* __restrict__ out, long long n)
{
    long long i = (long long)blockIdx.x * blockDim.x + threadIdx.x;
    if (i >= n) return;
    const int c = (int)(i % di);
    const long long t = i / di;
    const float z = xz[t * (size_t)(2 * di) + di + c];
    out[i] = y_in[i] * dev_silu(z);
}

__global__ void k_fuse_ws(const float* __restrict__ mix, const float* __restrict__ spk,
                          const float* __restrict__ a2, int C,
                          float* __restrict__ out, long long n)
{
    long long i = (long long)blockIdx.x * blockDim.x + threadIdx.x;
    if (i >= n) return;
    const long long t = i / C;
    const float w0 = a2[t * 2], w1 = a2[t * 2 + 1];
    const float mm = fmaxf(w0, w1);
    const float e0 = __expf(w0 - mm), e1 = __expf(w1 - mm);
    const float r = 1.f / (e0 + e1);
    out[i] = mix[i] * (e0 * r) + spk[i] * (e1 * r);
}

__global__ void k_gate_mix(const float* __restrict__ g, const float* __restrict__ sp,
                           const float* __restrict__ xp, float* __restrict__ y, long long n)
{
    long long i = (long long)blockIdx.x * blockDim.x + threadIdx.x;
    if (i >= n) return;
    const float gv = g[i];
    y[i] = gv * sp[i] + (1.f - gv) * xp[i];
}

// ============================================================================
// Host side
// ============================================================================

struct BnP  { const float *b, *g, *m, *v; };
struct BlkP { const float *A_log, *D, *conv_b, *conv_w, *dt_b, *dt_w,
                          *in_proj, *norm_b, *norm_g, *out_proj, *x_proj; };
struct DownP { BlkP blk[2]; BnP bn; const float *conv_b, *conv_w, *ln_b, *ln_g; };
struct FusP  { const float *a1_b, *a1_w, *a2_b, *a2_w; BnP bn1, bn2;
               const float *f1_b, *f1_w, *f2_b, *f2_w, *ln_b, *ln_g; };
struct UpP   { BlkP blk[2]; BnP bn1, bn2; const float *ct_b, *ct_w, *fu_b, *fu_w,
                           *g1_b, *g1_w, *g2_b, *g2_w, *gln_b, *gln_g, *ln_b, *ln_g; };
struct AttnP { const float *k_b, *k_w, *kn_b, *kn_g, *o_b, *o_w, *pos,
                           *q_b, *q_w, *qn_b, *qn_g, *v_b, *v_w; };

struct Arena {
    char* base; size_t off, cap;
    float* alloc(size_t nelem) {
        size_t bytes = (nelem * sizeof(float) + 255) & ~(size_t)255;
        float* p = (float*)(base + off);
        off += bytes;
        return p;
    }
};

static inline dim3 ewgrid(long long n) { return dim3((unsigned)((n + 255) / 256)); }

static void gemm(hipStream_t st, const float* X, int lda, const float* W, int ldb,
                 float* Y, int ldc, const float* bias, int M, int N, int K,
                 int act, bool transB)
{
    dim3 grid((N + 63) / 64, (M + 31) / 32), blk(128);
    const bool alignX = ((lda & 1) == 0) && (((uintptr_t)X & 7) == 0);
    const bool alignW = ((ldb & 1) == 0) && (((uintptr_t)W & 7) == 0);
    if (transB) {
        if (alignX && alignW)
            k_gemm<true, true ><<<grid, blk, 0, st>>>(X, lda, W, ldb, Y, ldc, bias, M, N, K, act);
        else
            k_gemm<true, false><<<grid, blk, 0, st>>>(X, lda, W, ldb, Y, ldc, bias, M, N, K, act);
    } else {
        if (alignX)
            k_gemm<false, true ><<<grid, blk, 0, st>>>(X, lda, W, ldb, Y, ldc, bias, M, N, K, act);
        else
            k_gemm<false, false><<<grid, blk, 0, st>>>(X, lda, W, ldb, Y, ldc, bias, M, N, K, act);
    }
}

// ------------------------------- Mamba -------------------------------------
static void run_mamba(hipStream_t st, Arena& A, const float* xln, float* out,
                      int Bn, int L, int d, const BlkP& p)
{
    const int di = 2 * d, dr = (d + 15) / 16, ndbl = dr + 32, M = Bn * L;
    const size_t mark = A.off;
    float* xz  = A.alloc((size_t)M * 2 * di);
    float* xc  = A.alloc((size_t)M * di);
    float* xd  = A.alloc((size_t)M * ndbl);
    float* dtp = A.alloc((size_t)M * di);
    float* yb  = A.alloc((size_t)M * di);
    float* yz  = A.alloc((size_t)M * di);
    const long long n1 = (long long)M * di;

    gemm(st, xln, d, p.in_proj, d, xz, 2 * di, nullptr, M, 2 * di, d, 0, true);
    k_dwconv_silu<<<ewgrid(n1), 256, 0, st>>>(xz, 2 * di, p.conv_w, p.conv_b, xc, L, di, Bn);
    gemm(st, xc, di, p.x_proj, di, xd, ndbl, nullptr, M, ndbl, di, 0, true);
    gemm(st, xd, ndbl, p.dt_w, dr, dtp, di, p.dt_b, M, di, dr, 0, true);
    k_scan<<<dim3((unsigned)Bn), dim3((unsigned)di), 0, st>>>(dtp, xd, xc, p.A_log, p.D, yb, L, di, ndbl, dr);
    k_mul_silu<<<ewgrid(n1), 256, 0, st>>>(yb, xz, di, yz, n1);
    gemm(st, yz, di, p.out_proj, di, out, d, nullptr, M, d, di, 0, true);
    A.off = mark;
}

static void run_mamba_layer(hipStream_t st, Arena& A, const float* x, float* out,
                            int Bn, int L, int d, const BlkP* blks)
{
    const int M = Bn * L;
    const long long n = (long long)M * d;
    const size_t mark = A.off;
    float* res = A.alloc((size_t)n);
    float* h   = A.alloc((size_t)n);
    float* lnb = A.alloc((size_t)n);

    k_copy<<<ewgrid(n), 256, 0, st>>>(x, res, n);
    for (int i = 0; i < 2; ++i) {
        if (i > 0) k_add<<<ewgrid(n), 256, 0, st>>>(res, h, res, n);
        k_layernorm<<<dim3((unsigned)M), 256, 0, st>>>(res, lnb, blks[i].norm_g, blks[i].norm_b, d, 0);
        run_mamba(st, A, lnb, h, Bn, L, d, blks[i]);
    }
    k_add<<<ewgrid(n), 256, 0, st>>>(h, res, out, n);
    A.off = mark;
}

// ------------------------------- Down --------------------------------------
static void run_down(hipStream_t st, Arena& A, const float* x, float* out,
                     int Bn, int Lin, int Cin, int Lout, int Cout, const DownP& p)
{
    const size_t mark = A.off;
    const long long n = (long long)Bn * Lout * Cout;
    float* c = A.alloc((size_t)n);
    float* m = A.alloc((size_t)n);
    k_conv_down<<<ewgrid(n), 256, 0, st>>>(x, Lin, Cin, p.conv_w, p.conv_b,
                                           p.bn.g, p.bn.b, p.bn.m, p.bn.v,
                                           c, Lout, Cout, Bn);
    run_mamba_layer(st, A, c, m, Bn, Lout, Cout, p.blk);
    k_layernorm<<<dim3((unsigned)(Bn * Lout)), 256, 0, st>>>(m, out, p.ln_g, p.ln_b, Cout, 0);
    A.off = mark;
}

// ---------------------------- Cross attention -------------------------------
static void run_attn(hipStream_t st, Arena& A, const float* e3m, const float* e3r,
                     float* spk, int Bn, int L3, const AttnP& p)
{
    const int C = 512, NH = 8, HD = 64, M = Bn * L3;
    const long long n = (long long)M * C;
    const size_t mark = A.off;
    float* qp = A.alloc((size_t)n);
    float* kp = A.alloc((size_t)n);
    float* qn = A.alloc((size_t)n);
    float* kn = A.alloc((size_t)n);
    float* Q  = A.alloc((size_t)n);
    float* Kb = A.alloc((size_t)n);
    float* V  = A.alloc((size_t)n);
    float* O  = A.alloc((size_t)n);
    float* att = A.alloc((size_t)n);
    float* sc = A.alloc((size_t)Bn * NH * L3 * L3);

    k_pos_add<<<ewgrid(n), 256, 0, st>>>(e3m, p.pos, qp, L3, C, n);
    k_pos_add<<<ewgrid(n), 256, 0, st>>>(e3r, p.pos, kp, L3, C, n);
    k_layernorm<<<dim3((unsigned)M), 256, 0, st>>>(qp, qn, p.qn_g, p.qn_b, C, 0);
    k_layernorm<<<dim3((unsigned)M), 256, 0, st>>>(kp, kn, p.kn_g, p.kn_b, C, 0);
    gemm(st, qn, C, p.q_w, C, Q,  C, p.q_b, M, C, C, 0, true);
    gemm(st, kn, C, p.k_w, C, Kb, C, p.k_b, M, C, C, 0, true);
    gemm(st, kn, C, p.v_w, C, V,  C, p.v_b, M, C, C, 0, true);

    for (int b = 0; b < Bn; ++b)
        for (int h = 0; h < NH; ++h) {
            const size_t qo = (size_t)b * L3 * C + (size_t)h * HD;
            float* sbh = sc + (size_t)(b * NH + h) * L3 * L3;
            // scores = Q_bh @ K_bh^T  (both row stride C)
            gemm(st, Q + qo, C, Kb + qo, C, sbh, L3, nullptr, L3, L3, HD, 0, true);
        }
    k_softmax_row<<<dim3((unsigned)(Bn * NH * L3)), 256, 0, st>>>(sc, L3, 0.125f);
    for (int b = 0; b < Bn; ++b)
        for (int h = 0; h < NH; ++h) {
            const size_t qo = (size_t)b * L3 * C + (size_t)h * HD;
            float* sbh = sc + (size_t)(b * NH + h) * L3 * L3;
            // O_bh = aw @ V_bh  (non-transposed B, row stride C)
            gemm(st, sbh, L3, V + qo, C, O + qo, C, nullptr, L3, HD, L3, 0, false);
        }
    gemm(st, O, C, p.o_w, C, att, C, p.o_b, M, C, C, 0, true);
    k_add<<<ewgrid(n), 256, 0, st>>>(att, qn, spk, n);
    A.off = mark;
}

// -------------------------------- Fusion ------------------------------------
static void run_fusion(hipStream_t st, Arena& A, const float* mix, const float* spk,
                       float* out, int Bn, int L3, const FusP& p)
{
    const int C = 512, M = Bn * L3;
    const long long n = (long long)M * C;
    const size_t mark = A.off;
    float* cat = A.alloc((size_t)M * 2 * C);
    float* a1  = A.alloc((size_t)n);
    float* a2  = A.alloc((size_t)M * 2);
    float* fw  = A.alloc((size_t)n);
    float* f1  = A.alloc((size_t)n);
    float* f2  = A.alloc((size_t)n);
    float* sum = A.alloc((size_t)n);

    k_concat2<<<ewgrid((long long)M * 2 * C), 256, 0, st>>>(mix, C, spk, C, cat, M);
    gemm(st, cat, 2 * C, p.a1_w, 2 * C, a1, C, p.a1_b, M, C, 2 * C, 1, true);  // silu
    gemm(st, a1, C, p.a2_w, C, a2, 2, p.a2_b, M, 2, C, 0, true);
    k_fuse_ws<<<ewgrid(n), 256, 0, st>>>(mix, spk, a2, C, fw, n);
    gemm(st, cat, 2 * C, p.f1_w, 2 * C, f1, C, p.f1_b, M, C, 2 * C, 0, true);
    k_bn_act<<<ewgrid(n), 256, 0, st>>>(f1, f1, p.bn1.g, p.bn1.b, p.bn1.m, p.bn1.v, C, n, 1);
    gemm(st, f1, C, p.f2_w, C, f2, C, p.f2_b, M, C, C, 0, true);
    k_bn_act<<<ewgrid(n), 256, 0, st>>>(f2, f2, p.bn2.g, p.bn2.b, p.bn2.m, p.bn2.v, C, n, 0);
    k_add<<<ewgrid(n), 256, 0, st>>>(fw, f2, sum, n);
    k_layernorm<<<dim3((unsigned)M), 256, 0, st>>>(sum, out, p.ln_g, p.ln_b, C, 0);
    A.off = mark;
}

// ---------------------------------- Up --------------------------------------
static void run_up(hipStream_t st, Arena& A, const float* x, int Lin, int Cin,
                   const float* skip, int Lsk, int Cout, float* out,
                   int Bn, const UpP& p)
{
    const int M = Bn * Lsk;
    const long long n = (long long)M * Cout;
    const size_t mark = A.off;
    float* xt   = A.alloc((size_t)n);
    float* cat  = A.alloc((size_t)M * 2 * Cout);
    float* x0   = A.alloc((size_t)n);
    float* xm   = A.alloc((size_t)n);
    float* cat2 = A.alloc((size_t)M * 2 * Cout);
    float* g1o  = A.alloc((size_t)n);
    float* g    = A.alloc((size_t)n);
    float* mixb = A.alloc((size_t)n);

    k_convT<<<ewgrid(n), 256, 0, st>>>(x, Lin, Cin, p.ct_w, p.ct_b,
                                       p.bn1.g, p.bn1.b, p.bn1.m, p.bn1.v, 1,
                                       xt, Lsk, Cout, Bn);
    k_concat2<<<ewgrid((long long)M * 2 * Cout), 256, 0, st>>>(skip, Cout, xt, Cout, cat, M);
    gemm(st, cat, 2 * Cout, p.fu_w, 2 * Cout, x0, Cout, p.fu_b, M, Cout, 2 * Cout, 0, true);
    k_bn_act<<<ewgrid(n), 256, 0, st>>>(x0, x0, p.bn2.g, p.bn2.b, p.bn2.m, p.bn2.v, Cout, n, 1);
    run_mamba_layer(st, A, x0, xm, Bn, Lsk, Cout, p.blk);
    k_concat2<<<ewgrid((long long)M * 2 * Cout), 256, 0, st>>>(xm, Cout, x0, Cout, cat2, M);
    gemm(st, cat2, 2 * Cout, p.g1_w, 2 * Cout, g1o, Cout, p.g1_b, M, Cout, 2 * Cout, 0, true);
    k_layernorm<<<dim3((unsigned)M), 256, 0, st>>>(g1o, g1o, p.gln_g, p.gln_b, Cout, 1); // silu(LN)
    gemm(st, g1o, Cout, p.g2_w, Cout, g, Cout, p.g2_b, M, Cout, Cout, 2, true);          // sigmoid
    k_gate_mix<<<ewgrid(n), 256, 0, st>>>(g, x0, xm, mixb, n);
    k_layernorm<<<dim3((unsigned)M), 256, 0, st>>>(mixb, out, p.ln_g, p.ln_b, Cout, 0);
    A.off = mark;
}

// ============================================================================
// kernel_launch
// ============================================================================
extern "C" void kernel_launch(void* const* d_in, const int* in_sizes, int n_in,
                              void* d_out, int out_size, void* d_ws, size_t ws_size,
                              hipStream_t stream)
{
    (void)in_sizes; (void)n_in; (void)out_size;
    const int Bn = 2, T = 8000;
    const int L1 = 4001, L2 = 2001, L3 = 1001;
    const int C1 = 128, C2 = 256, C3 = 512;

    // ---- parse inputs (jax tree flatten order: dict keys sorted, DFS) ----
    const float* mixture = (const float*)d_in[0];
    const float* refwav  = (const float*)d_in[1];
    struct Cursor { void* const* d; int i; const float* nx() { return (const float*)d[i++]; } };
    Cursor cu{d_in, 2};

    AttnP ap;
    ap.k_b = cu.nx(); ap.k_w = cu.nx(); ap.kn_b = cu.nx(); ap.kn_g = cu.nx();
    ap.o_b = cu.nx(); ap.o_w = cu.nx(); ap.pos = cu.nx();
    ap.q_b = cu.nx(); ap.q_w = cu.nx(); ap.qn_b = cu.nx(); ap.qn_g = cu.nx();
    ap.v_b = cu.nx(); ap.v_w = cu.nx();

    auto parseBlk = [&](BlkP& b) {
        b.A_log = cu.nx(); b.D = cu.nx(); b.conv_b = cu.nx(); b.conv_w = cu.nx();
        b.dt_b = cu.nx(); b.dt_w = cu.nx(); b.in_proj = cu.nx();
        b.norm_b = cu.nx(); b.norm_g = cu.nx(); b.out_proj = cu.nx(); b.x_proj = cu.nx();
    };
    auto parseBn = [&](BnP& b) { b.b = cu.nx(); b.g = cu.nx(); b.m = cu.nx(); b.v = cu.nx(); };

    DownP dp[3];
    for (int i = 0; i < 3; ++i) {
        parseBlk(dp[i].blk[0]); parseBlk(dp[i].blk[1]);
        parseBn(dp[i].bn);
        dp[i].conv_b = cu.nx(); dp[i].conv_w = cu.nx();
        dp[i].ln_b = cu.nx(); dp[i].ln_g = cu.nx();
    }

    FusP fp;
    fp.a1_b = cu.nx(); fp.a1_w = cu.nx(); fp.a2_b = cu.nx(); fp.a2_w = cu.nx();
    parseBn(fp.bn1); parseBn(fp.bn2);
    fp.f1_b = cu.nx(); fp.f1_w = cu.nx(); fp.f2_b = cu.nx(); fp.f2_w = cu.nx();
    fp.ln_b = cu.nx(); fp.ln_g = cu.nx();

    const float* out_b = cu.nx();
    const float* out_w = cu.nx();

    UpP up[2];
    for (int i = 0; i < 2; ++i) {
        parseBlk(up[i].blk[0]); parseBlk(up[i].blk[1]);
        parseBn(up[i].bn1); parseBn(up[i].bn2);
        up[i].ct_b = cu.nx(); up[i].ct_w = cu.nx();
        up[i].fu_b = cu.nx(); up[i].fu_w = cu.nx();
        up[i].g1_b = cu.nx(); up[i].g1_w = cu.nx();
        up[i].g2_b = cu.nx(); up[i].g2_w = cu.nx();
        up[i].gln_b = cu.nx(); up[i].gln_g = cu.nx();
        up[i].ln_b = cu.nx(); up[i].ln_g = cu.nx();
    }

    // ---- workspace arena: persistent buffers first ----
    Arena A{(char*)d_ws, 0, ws_size};
    float* e1m = A.alloc((size_t)Bn * L1 * C1);
    float* e2m = A.alloc((size_t)Bn * L2 * C2);
    float* e3m = A.alloc((size_t)Bn * L3 * C3);
    float* e1r = A.alloc((size_t)Bn * L1 * C1);
    float* e2r = A.alloc((size_t)Bn * L2 * C2);
    float* e3r = A.alloc((size_t)Bn * L3 * C3);
    float* spk = A.alloc((size_t)Bn * L3 * C3);
    float* fus = A.alloc((size_t)Bn * L3 * C3);
    float* u1  = A.alloc((size_t)Bn * L2 * C2);
    float* u2  = A.alloc((size_t)Bn * L1 * C1);

    // ---- encoder (mixture + reference share down params) ----
    run_down(stream, A, mixture, e1m, Bn, T,  1,  L1, C1, dp[0]);
    run_down(stream, A, e1m,     e2m, Bn, L1, C1, L2, C2, dp[1]);
    run_down(stream, A, e2m,     e3m, Bn, L2, C2, L3, C3, dp[2]);
    run_down(stream, A, refwav,  e1r, Bn, T,  1,  L1, C1, dp[0]);
    run_down(stream, A, e1r,     e2r, Bn, L1, C1, L2, C2, dp[1]);
    run_down(stream, A, e2r,     e3r, Bn, L2, C2, L3, C3, dp[2]);

    // ---- cross attention + fusion ----
    run_attn(stream, A, e3m, e3r, spk, Bn, L3, ap);
    run_fusion(stream, A, e3m, spk, fus, Bn, L3, fp);

    // ---- decoder ----
    run_up(stream, A, fus, L3, C3, e2m, L2, C2, u1, Bn, up[0]);
    run_up(stream, A, u1,  L2, C2, e1m, L1, C1, u2, Bn, up[1]);

    // ---- output convT (128 -> 1), truncated to T ----
    float* dout = (float*)d_out;
    const long long nout = (long long)Bn * T;
    k_convT<<<ewgrid(nout), 256, 0, stream>>>(u2, L1, C1, out_w, out_b,
                                              nullptr, nullptr, nullptr, nullptr, 0,
                                              dout, T, 1, Bn);
}